// SMPL2MeshOptimizer_51204600103495
// MI455X (gfx1250) — compile-verified
//
#include <hip/hip_runtime.h>
#include <math.h>

// ---------------------------------------------------------------------------
// MI455X mesh-loss kernel.
// Chamfer terms (~1.6e9 point pairs) dominate and are pure compute (all point
// sets < 1 MB, trivially L2-resident vs 23.3 TB/s HBM). They are mapped onto
// V_WMMA_F32_16X16X4_F32: d(i,j) = |a|^2 + |b|^2 - 2 a.b computed as
//   D = A(16x4) x B(4x16) + C,  A=[-2ax,-2ay,-2az,1], B=[bx,by,bz,|b|^2]^T,
//   C = |a|^2 broadcast along N  -> one WMMA per 256 complete distances.
// B is pre-split into (x,y)/(z,|b|^2) halves matching the half-wave K layout
// so the inner loop is ds_load_b64 + wmma + 8x v_min (no per-tile selects),
// and B chunks are double-buffered into LDS with GLOBAL_LOAD_ASYNC_TO_LDS
// (ASYNCcnt) when the toolchain exposes the builtins.
// ---------------------------------------------------------------------------

typedef float v2f __attribute__((ext_vector_type(2)));
typedef float v8f __attribute__((ext_vector_type(8)));

#define NBLK      256    // grid size for grid-stride elementwise kernels
#define PSTRIDE   256    // partial entries per loss term
#define EPSF      1e-6f
#define CHUNK_PTS 1024   // B points per LDS buffer (8 KB Lo + 8 KB Hi)

#if defined(__has_builtin)
#  if __has_builtin(__builtin_amdgcn_global_load_async_to_lds_b128) && \
      __has_builtin(__builtin_amdgcn_s_wait_asynccnt)
#    define USE_ASYNC_LDS 1
#  endif
#endif
#ifndef USE_ASYNC_LDS
#  define USE_ASYNC_LDS 0
#endif

typedef int v4i_vs __attribute__((vector_size(4 * sizeof(int))));
typedef __attribute__((address_space(1))) v4i_vs* g_v4i_ptr;
typedef __attribute__((address_space(3))) v4i_vs* l_v4i_ptr;

__device__ __forceinline__ void copy_b128(const void* g, void* l) {
#if USE_ASYNC_LDS
  __builtin_amdgcn_global_load_async_to_lds_b128(
      (g_v4i_ptr)g, (l_v4i_ptr)l, 0, 0);
#else
  *(float4*)l = *(const float4*)g;
#endif
}
__device__ __forceinline__ void copy_wait() {
#if USE_ASYNC_LDS
  __builtin_amdgcn_s_wait_asynccnt(0);
#endif
}

__device__ __forceinline__ float block_sum256(float v, float* sm) {
  int t = threadIdx.x;
  sm[t] = v;
  __syncthreads();
  for (int s = 128; s > 0; s >>= 1) {
    if (t < s) sm[t] += sm[t + s];
    __syncthreads();
  }
  float r = sm[0];
  __syncthreads();
  return r;
}

// ---- prep: new_verts, flip, split B layouts, penetration (slot 12) --------
__global__ __launch_bounds__(256) void prep_verts_kernel(
    const float* __restrict__ verts, const float* __restrict__ deform,
    const float* __restrict__ vnormal,
    float4* __restrict__ nv4, float4* __restrict__ flip4,
    float2* __restrict__ nvLo, float2* __restrict__ nvHi,
    float2* __restrict__ flipLo, float2* __restrict__ flipHi,
    float* __restrict__ partials, int V) {
  __shared__ float sm[256];
  float pen = 0.f;
  for (int v = blockIdx.x * blockDim.x + threadIdx.x; v < V;
       v += gridDim.x * blockDim.x) {
    float x = verts[3*v+0] + deform[3*v+0];
    float y = verts[3*v+1] + deform[3*v+1];
    float z = verts[3*v+2] + deform[3*v+2];
    float sq = x*x + y*y + z*z;
    nv4[v]     = make_float4( x, y, z, sq);
    flip4[v]   = make_float4(-x, y, z, sq);
    nvLo[v]    = make_float2( x, y);
    nvHi[v]    = make_float2( z, sq);
    flipLo[v]  = make_float2(-x, y);
    flipHi[v]  = make_float2( z, sq);
    float d = deform[3*v+0]*vnormal[3*v+0] + deform[3*v+1]*vnormal[3*v+1]
            + deform[3*v+2]*vnormal[3*v+2];
    pen += fabsf(fminf(d, -0.1f));
  }
  float s = block_sum256(pen, sm);
  if (threadIdx.x == 0) partials[12*PSTRIDE + blockIdx.x] = s;
}

__global__ __launch_bounds__(256) void prep_trg_kernel(
    const float* __restrict__ trg, float4* __restrict__ trg4,
    float2* __restrict__ trgLo, float2* __restrict__ trgHi, int Nt) {
  for (int t = blockIdx.x * blockDim.x + threadIdx.x; t < Nt;
       t += gridDim.x * blockDim.x) {
    float x = trg[3*t], y = trg[3*t+1], z = trg[3*t+2];
    float sq = x*x + y*y + z*z;
    trg4[t]  = make_float4(x, y, z, sq);
    trgLo[t] = make_float2(x, y);
    trgHi[t] = make_float2(z, sq);
  }
}

__global__ __launch_bounds__(256) void sample_src_kernel(
    const float4* __restrict__ nv4, const int* __restrict__ faces,
    const int* __restrict__ sfidx, const float* __restrict__ bary,
    float4* __restrict__ src4, float2* __restrict__ srcLo,
    float2* __restrict__ srcHi, int Ns) {
  for (int n = blockIdx.x * blockDim.x + threadIdx.x; n < Ns;
       n += gridDim.x * blockDim.x) {
    int f = sfidx[n];
    int i0 = faces[3*f], i1 = faces[3*f+1], i2 = faces[3*f+2];
    float4 a = nv4[i0], b = nv4[i1], c = nv4[i2];
    float b0 = bary[3*n], b1 = bary[3*n+1], b2 = bary[3*n+2];
    float x = b0*a.x + b1*b.x + b2*c.x;
    float y = b0*a.y + b1*b.y + b2*c.y;
    float z = b0*a.z + b1*b.z + b2*c.z;
    float sq = x*x + y*y + z*z;
    src4[n]  = make_float4(x, y, z, sq);
    srcLo[n] = make_float2(x, y);
    srcHi[n] = make_float2(z, sq);
  }
}

// ---- chamfer core: per-row min squared distance via f32 WMMA ---------------
// Block = 256 thr = 8 waves, 16 rows/wave (128 rows/block). gridDim.y splits
// the B columns in halves for occupancy; out_min[y*Na + row] holds the split
// partial min. Na % 128 == 0, NbHalf % 1024 == 0.
__global__ __launch_bounds__(256) void chamfer_min_kernel(
    const float4* __restrict__ A4,
    const float2* __restrict__ BLo, const float2* __restrict__ BHi,
    int NbHalf, int Na, float* __restrict__ out_min) {
  __shared__ __align__(16) float2 sLo[2][CHUNK_PTS];
  __shared__ __align__(16) float2 sHi[2][CHUNK_PTS];

  const int tid  = threadIdx.x;
  const int lane = tid & 31;
  const int l15  = lane & 15;
  const bool hi  = lane >= 16;
  const int rowBase = blockIdx.x * 128 + (tid >> 5) * 16;
  const int colBase = blockIdx.y * NbHalf;

  // A fragment (16x4 f32): lanes 0-15 -> K0,K1 ; lanes 16-31 -> K2,K3
  float4 ap = A4[rowBase + l15];
  v2f afrag;
  afrag.x = hi ? (-2.0f * ap.z) : (-2.0f * ap.x);
  afrag.y = hi ? 1.0f            : (-2.0f * ap.y);

  // C fragment: c[r] = |a|^2 of row (r + 8*hi), broadcast along N
  v8f cfrag;
#pragma unroll
  for (int r = 0; r < 8; ++r)
    cfrag[r] = A4[rowBase + (hi ? 8 : 0) + r].w;

  // per-lane LDS base pointers (half-wave select done once, not per tile)
  const float2* p0 = hi ? sHi[0] : sLo[0];
  const float2* p1 = hi ? sHi[1] : sLo[1];

  v8f mins;
#pragma unroll
  for (int r = 0; r < 8; ++r) mins[r] = 3.402823e38f;

  // prologue: stage chunk 0 into buffer 0 (each thread moves 2x b128/array)
  {
    const float2* gLo = BLo + colBase;
    const float2* gHi = BHi + colBase;
#pragma unroll
    for (int k = 0; k < 2; ++k) {
      int idx = 2 * (tid + k * 256);       // float2 index, 16B granules
      copy_b128(gLo + idx, &sLo[0][idx]);
      copy_b128(gHi + idx, &sHi[0][idx]);
    }
    copy_wait();
    __syncthreads();
  }

  int cur = 0;
  for (int chunk = 0; chunk < NbHalf; chunk += CHUNK_PTS) {
    int nxt = chunk + CHUNK_PTS;
    if (nxt < NbHalf) {                    // block-uniform branch
      const float2* gLo = BLo + colBase + nxt;
      const float2* gHi = BHi + colBase + nxt;
      float2* dLo = sLo[cur ^ 1];
      float2* dHi = sHi[cur ^ 1];
#pragma unroll
      for (int k = 0; k < 2; ++k) {
        int idx = 2 * (tid + k * 256);
        copy_b128(gLo + idx, dLo + idx);
        copy_b128(gHi + idx, dHi + idx);
      }
    }
    const float2* p = cur ? p1 : p0;
#pragma unroll 8
    for (int j = 0; j < CHUNK_PTS; j += 16) {
      float2 bv = p[j + l15];              // ds_load_b64
      v2f bfrag;
      bfrag.x = bv.x;
      bfrag.y = bv.y;
      v8f d = __builtin_amdgcn_wmma_f32_16x16x4_f32(
          false, afrag, false, bfrag, 0, cfrag, false, false);
#pragma unroll
      for (int r = 0; r < 8; ++r) mins[r] = fminf(mins[r], d[r]);
    }
    copy_wait();
    __syncthreads();
    cur ^= 1;
  }

  // min across the 16 lanes of each half-wave (N dimension)
#pragma unroll
  for (int r = 0; r < 8; ++r) {
    float m = mins[r];
    m = fminf(m, __shfl_xor(m, 8, 32));
    m = fminf(m, __shfl_xor(m, 4, 32));
    m = fminf(m, __shfl_xor(m, 2, 32));
    m = fminf(m, __shfl_xor(m, 1, 32));
    mins[r] = m;
  }
  if (l15 == 0) {
    int base = blockIdx.y * Na + rowBase + (hi ? 8 : 0);
#pragma unroll
    for (int r = 0; r < 8; ++r) out_min[base + r] = mins[r];
  }
}

// ---- sum of min over the two column splits into a partial slot ------------
__global__ __launch_bounds__(256) void reduce_minsum_kernel(
    const float* __restrict__ x, int n, float* __restrict__ partials, int slot) {
  __shared__ float sm[256];
  float s = 0.f;
  for (int i = blockIdx.x * blockDim.x + threadIdx.x; i < n;
       i += gridDim.x * blockDim.x)
    s += fminf(x[i], x[n + i]);
  s = block_sum256(s, sm);
  if (threadIdx.x == 0) partials[slot*PSTRIDE + blockIdx.x] = s;
}

// ---- face pass A: edge lengths, pairwise L1 terms, smooth, face normals ----
__global__ __launch_bounds__(256) void faceA_kernel(
    const float4* __restrict__ nv4, const int* __restrict__ faces,
    const float* __restrict__ deform,
    float* __restrict__ e0a, float* __restrict__ e1a, float* __restrict__ e2a,
    float4* __restrict__ fnu4, float* __restrict__ vsum,
    float* __restrict__ partials, int F) {
  __shared__ float sm[256];
  float s_sum = 0.f, s_pair = 0.f, s_smooth = 0.f;
  for (int f = blockIdx.x * blockDim.x + threadIdx.x; f < F;
       f += gridDim.x * blockDim.x) {
    int i0 = faces[3*f], i1 = faces[3*f+1], i2 = faces[3*f+2];
    float4 v0 = nv4[i0], v1 = nv4[i1], v2 = nv4[i2];
    float e0 = sqrtf((v0.x-v1.x)*(v0.x-v1.x)+(v0.y-v1.y)*(v0.y-v1.y)+(v0.z-v1.z)*(v0.z-v1.z));
    float e1 = sqrtf((v1.x-v2.x)*(v1.x-v2.x)+(v1.y-v2.y)*(v1.y-v2.y)+(v1.z-v2.z)*(v1.z-v2.z));
    float e2 = sqrtf((v2.x-v0.x)*(v2.x-v0.x)+(v2.y-v0.y)*(v2.y-v0.y)+(v2.z-v0.z)*(v2.z-v0.z));
    e0a[f] = e0; e1a[f] = e1; e2a[f] = e2;
    s_sum  += e0 + e1 + e2;
    s_pair += fabsf(e0-e1) + fabsf(e1-e2) + fabsf(e2-e0);
    float s0 = sqrtf(deform[3*i0]*deform[3*i0]+deform[3*i0+1]*deform[3*i0+1]+deform[3*i0+2]*deform[3*i0+2]);
    float s1 = sqrtf(deform[3*i1]*deform[3*i1]+deform[3*i1+1]*deform[3*i1+1]+deform[3*i1+2]*deform[3*i1+2]);
    float s2 = sqrtf(deform[3*i2]*deform[3*i2]+deform[3*i2+1]*deform[3*i2+1]+deform[3*i2+2]*deform[3*i2+2]);
    s_smooth += fabsf(s0-s1) + fabsf(s1-s2) + fabsf(s2-s0);
    float ax = v1.x-v0.x, ay = v1.y-v0.y, az = v1.z-v0.z;
    float bx = v2.x-v0.x, by = v2.y-v0.y, bz = v2.z-v0.z;
    float nx = ay*bz - az*by, ny = az*bx - ax*bz, nz = ax*by - ay*bx;
    float inv = 1.f / (sqrtf(nx*nx+ny*ny+nz*nz) + EPSF);
    fnu4[f] = make_float4(nx*inv, ny*inv, nz*inv, 0.f);
    atomicAdd(vsum + 4*i0 + 0, nx); atomicAdd(vsum + 4*i0 + 1, ny); atomicAdd(vsum + 4*i0 + 2, nz);
    atomicAdd(vsum + 4*i1 + 0, nx); atomicAdd(vsum + 4*i1 + 1, ny); atomicAdd(vsum + 4*i1 + 2, nz);
    atomicAdd(vsum + 4*i2 + 0, nx); atomicAdd(vsum + 4*i2 + 1, ny); atomicAdd(vsum + 4*i2 + 2, nz);
  }
  float a = block_sum256(s_sum,    sm);
  if (threadIdx.x == 0) partials[4*PSTRIDE + blockIdx.x] = a;
  float b = block_sum256(s_pair,   sm);
  if (threadIdx.x == 0) partials[5*PSTRIDE + blockIdx.x] = b;
  float c = block_sum256(s_smooth, sm);
  if (threadIdx.x == 0) partials[6*PSTRIDE + blockIdx.x] = c;
}

__global__ __launch_bounds__(256) void edge_scatter_kernel(
    const float4* __restrict__ nv4, const int* __restrict__ edges,
    float* __restrict__ nbr, float* __restrict__ deg, int E) {
  for (int e = blockIdx.x * blockDim.x + threadIdx.x; e < E;
       e += gridDim.x * blockDim.x) {
    int a = edges[2*e], b = edges[2*e+1];
    float4 va = nv4[a], vb = nv4[b];
    atomicAdd(nbr + 4*a + 0, vb.x); atomicAdd(nbr + 4*a + 1, vb.y); atomicAdd(nbr + 4*a + 2, vb.z);
    atomicAdd(nbr + 4*b + 0, va.x); atomicAdd(nbr + 4*b + 1, va.y); atomicAdd(nbr + 4*b + 2, va.z);
    atomicAdd(deg + a, 1.f);
    atomicAdd(deg + b, 1.f);
  }
}

__global__ void avg_len_kernel(const float* __restrict__ partials,
                               float* __restrict__ scal, int F) {
  float s = 0.f;
  for (int i = 0; i < PSTRIDE; ++i) s += partials[4*PSTRIDE + i];
  scal[0] = s / (3.0f * (float)F);
}

__global__ __launch_bounds__(256) void faceB_kernel(
    const float* __restrict__ e0a, const float* __restrict__ e1a,
    const float* __restrict__ e2a, const float* __restrict__ scal,
    float* __restrict__ partials, int F) {
  __shared__ float sm[256];
  float avg = scal[0];
  float s = 0.f;
  for (int f = blockIdx.x * blockDim.x + threadIdx.x; f < F;
       f += gridDim.x * blockDim.x)
    s += fabsf(e0a[f]-avg) + fabsf(e1a[f]-avg) + fabsf(e2a[f]-avg);
  float r = block_sum256(s, sm);
  if (threadIdx.x == 0) partials[7*PSTRIDE + blockIdx.x] = r;
}

__global__ __launch_bounds__(256) void edgeB_kernel(
    const float4* __restrict__ nv4, const int* __restrict__ edges,
    const float* __restrict__ scal, float* __restrict__ partials, int E) {
  __shared__ float sm[256];
  float avg = scal[0];
  float s = 0.f;
  for (int e = blockIdx.x * blockDim.x + threadIdx.x; e < E;
       e += gridDim.x * blockDim.x) {
    int a = edges[2*e], b = edges[2*e+1];
    float4 va = nv4[a], vb = nv4[b];
    float dx = va.x-vb.x, dy = va.y-vb.y, dz = va.z-vb.z;
    float elen = sqrtf(dx*dx + dy*dy + dz*dz);
    float d = elen - avg;
    s += d*d;
  }
  float r = block_sum256(s, sm);
  if (threadIdx.x == 0) partials[8*PSTRIDE + blockIdx.x] = r;
}

__global__ __launch_bounds__(256) void normal_kernel(
    const float4* __restrict__ fnu4, const int* __restrict__ fpairs,
    float* __restrict__ partials, int P) {
  __shared__ float sm[256];
  float s = 0.f;
  for (int p = blockIdx.x * blockDim.x + threadIdx.x; p < P;
       p += gridDim.x * blockDim.x) {
    float4 n0 = fnu4[fpairs[2*p]], n1 = fnu4[fpairs[2*p+1]];
    s += 1.0f - (n0.x*n1.x + n0.y*n1.y + n0.z*n1.z);
  }
  float r = block_sum256(s, sm);
  if (threadIdx.x == 0) partials[9*PSTRIDE + blockIdx.x] = r;
}

__global__ __launch_bounds__(256) void vertex_kernel(
    const float4* __restrict__ nv4, const float* __restrict__ nbr,
    const float* __restrict__ deg, const float* __restrict__ vsum,
    const float* __restrict__ vnormal, float* __restrict__ partials, int V) {
  __shared__ float sm[256];
  float s_lap = 0.f, s_con = 0.f;
  for (int v = blockIdx.x * blockDim.x + threadIdx.x; v < V;
       v += gridDim.x * blockDim.x) {
    float4 nv = nv4[v];
    float d = fmaxf(deg[v], 1.0f);
    float lx = nbr[4*v+0]/d - nv.x;
    float ly = nbr[4*v+1]/d - nv.y;
    float lz = nbr[4*v+2]/d - nv.z;
    s_lap += sqrtf(lx*lx + ly*ly + lz*lz);
    float vx = vsum[4*v+0], vy = vsum[4*v+1], vz = vsum[4*v+2];
    float inv = 1.f / (sqrtf(vx*vx + vy*vy + vz*vz) + EPSF);
    float dx = vx*inv - vnormal[3*v+0];
    float dy = vy*inv - vnormal[3*v+1];
    float dz = vz*inv - vnormal[3*v+2];
    s_con += dx*dx + dy*dy + dz*dz;
  }
  float a = block_sum256(s_lap, sm);
  if (threadIdx.x == 0) partials[10*PSTRIDE + blockIdx.x] = a;
  float b = block_sum256(s_con, sm);
  if (threadIdx.x == 0) partials[11*PSTRIDE + blockIdx.x] = b;
}

__global__ void combine_kernel(const float* __restrict__ partials,
                               float* __restrict__ out,
                               int Ns, int Nt, int V, int F, int E, int P) {
  float S[13];
  for (int k = 0; k < 13; ++k) {
    float s = 0.f;
    for (int i = 0; i < PSTRIDE; ++i) s += partials[k*PSTRIDE + i];
    S[k] = s;
  }
  float loss = 0.f;
  loss += 1.0f * (S[0]/(float)Ns + S[1]/(float)Nt);       // chamfer
  loss += 0.1f * ((S[5] + S[7])/(float)F);                 // face
  loss += 0.1f * (S[8]/(float)E);                          // edge
  loss += 0.1f * (S[9]/(float)P);                          // normal
  loss += 0.1f * (S[6]/(float)F);                          // smooth
  loss += 0.1f * (S[10]/(float)V);                         // laplacian
  loss += 0.5f * (S[11]/(3.0f*(float)V));                  // consistency
  loss += 0.1f * (S[2]/(float)V + S[3]/(float)V);          // symmetry
  loss += 5.0f * (S[12]/(float)V);                         // penetration
  out[0] = loss;
}

// ---------------------------------------------------------------------------
extern "C" void kernel_launch(void* const* d_in, const int* in_sizes, int n_in,
                              void* d_out, int out_size, void* d_ws, size_t ws_size,
                              hipStream_t stream) {
  (void)n_in; (void)out_size; (void)ws_size;
  const float* verts      = (const float*)d_in[0];
  const float* deform     = (const float*)d_in[1];
  const float* src_bary   = (const float*)d_in[2];
  const float* sample_trg = (const float*)d_in[3];
  const float* vnormal    = (const float*)d_in[4];
  const int*   faces      = (const int*)d_in[5];
  const int*   edges      = (const int*)d_in[6];
  const int*   fpairs     = (const int*)d_in[7];
  const int*   sfidx      = (const int*)d_in[8];

  const int V  = in_sizes[0] / 3;
  const int Ns = in_sizes[2] / 3;
  const int Nt = in_sizes[3] / 3;
  const int F  = in_sizes[5] / 3;
  const int E  = in_sizes[6] / 2;
  const int P  = in_sizes[7] / 2;

  float* w = (float*)d_ws;
  size_t off = 0;
  float4* nv4    = (float4*)(w + off); off += (size_t)V  * 4;
  float4* flip4  = (float4*)(w + off); off += (size_t)V  * 4;
  float4* src4   = (float4*)(w + off); off += (size_t)Ns * 4;
  float4* trg4   = (float4*)(w + off); off += (size_t)Nt * 4;
  float2* nvLo   = (float2*)(w + off); off += (size_t)V  * 2;
  float2* nvHi   = (float2*)(w + off); off += (size_t)V  * 2;
  float2* flipLo = (float2*)(w + off); off += (size_t)V  * 2;
  float2* flipHi = (float2*)(w + off); off += (size_t)V  * 2;
  float2* srcLo  = (float2*)(w + off); off += (size_t)Ns * 2;
  float2* srcHi  = (float2*)(w + off); off += (size_t)Ns * 2;
  float2* trgLo  = (float2*)(w + off); off += (size_t)Nt * 2;
  float2* trgHi  = (float2*)(w + off); off += (size_t)Nt * 2;
  float* min_src  = w + off; off += (size_t)Ns * 2;   // two column splits
  float* min_trg  = w + off; off += (size_t)Nt * 2;
  float* min_nv   = w + off; off += (size_t)V  * 2;
  float* min_flip = w + off; off += (size_t)V  * 2;
  float* e0a = w + off; off += F;
  float* e1a = w + off; off += F;
  float* e2a = w + off; off += F;
  float4* fnu4 = (float4*)(w + off); off += (size_t)F * 4;
  float* vsum = w + off; off += (size_t)V * 4;
  float* nbr  = w + off; off += (size_t)V * 4;
  float* degA = w + off; off += V;
  float* partials = w + off; off += 13 * PSTRIDE;
  float* scal = w + off; off += 4;

  // zero the scatter targets (vsum | nbr | deg are contiguous)
  (void)hipMemsetAsync(vsum, 0, (size_t)(9 * V) * sizeof(float), stream);

  prep_verts_kernel<<<NBLK, 256, 0, stream>>>(verts, deform, vnormal,
      nv4, flip4, nvLo, nvHi, flipLo, flipHi, partials, V);
  prep_trg_kernel<<<NBLK, 256, 0, stream>>>(sample_trg, trg4, trgLo, trgHi, Nt);
  sample_src_kernel<<<NBLK, 256, 0, stream>>>(nv4, faces, sfidx, src_bary,
                                              src4, srcLo, srcHi, Ns);

  // chamfer (WMMA): Na % 128 == 0, Nb % 2048 == 0 for these shapes
  dim3 g1(Ns / 128, 2), g2(Nt / 128, 2), g3(V / 128, 2);
  chamfer_min_kernel<<<g1, 256, 0, stream>>>(src4,  trgLo,  trgHi,  Nt/2, Ns, min_src);
  chamfer_min_kernel<<<g2, 256, 0, stream>>>(trg4,  srcLo,  srcHi,  Ns/2, Nt, min_trg);
  chamfer_min_kernel<<<g3, 256, 0, stream>>>(nv4,   flipLo, flipHi, V/2,  V,  min_nv);
  chamfer_min_kernel<<<g3, 256, 0, stream>>>(flip4, nvLo,   nvHi,   V/2,  V,  min_flip);

  reduce_minsum_kernel<<<NBLK, 256, 0, stream>>>(min_src,  Ns, partials, 0);
  reduce_minsum_kernel<<<NBLK, 256, 0, stream>>>(min_trg,  Nt, partials, 1);
  reduce_minsum_kernel<<<NBLK, 256, 0, stream>>>(min_nv,   V,  partials, 2);
  reduce_minsum_kernel<<<NBLK, 256, 0, stream>>>(min_flip, V,  partials, 3);

  faceA_kernel<<<NBLK, 256, 0, stream>>>(nv4, faces, deform, e0a, e1a, e2a,
                                         fnu4, vsum, partials, F);
  edge_scatter_kernel<<<NBLK, 256, 0, stream>>>(nv4, edges, nbr, degA, E);
  avg_len_kernel<<<1, 1, 0, stream>>>(partials, scal, F);
  faceB_kernel<<<NBLK, 256, 0, stream>>>(e0a, e1a, e2a, scal, partials, F);
  edgeB_kernel<<<NBLK, 256, 0, stream>>>(nv4, edges, scal, partials, E);
  normal_kernel<<<NBLK, 256, 0, stream>>>(fnu4, fpairs, partials, P);
  vertex_kernel<<<NBLK, 256, 0, stream>>>(nv4, nbr, degA, vsum, vnormal,
                                          partials, V);
  combine_kernel<<<1, 1, 0, stream>>>(partials, (float*)d_out,
                                      Ns, Nt, V, F, E, P);
}